// AttentionHead_21758304321960
// MI455X (gfx1250) — compile-verified
//
#include <hip/hip_runtime.h>

// CDNA5 / gfx1250 single-head causal attention, fp32 end-to-end.
//  Kernel 0: transpose Wq/Wk/Wv -> W^T so GEMM B-operands load as b64 pairs.
//  Kernel 1: QKV projection, V_WMMA_F32_16X16X4_F32, 64-row M-tile per wave,
//            one weight matrix per wave (grid.y).  V is stored transposed.
//  Kernel 2: flash attention, one wave32 per 16-query block, online softmax.
//            S^T = K*Q^T puts queries on lanes; P feeds PV WMMA as B operand
//            via half-swap shuffles; V^T makes the PV A-operand loads b64.

typedef float v2f __attribute__((ext_vector_type(2)));
typedef float v8f __attribute__((ext_vector_type(8)));

#define WMMA_F32(A, Bm, C) \
    __builtin_amdgcn_wmma_f32_16x16x4_f32(false, (A), false, (Bm), (short)0, (C), false, false)

constexpr int BB = 4;        // batch
constexpr int SS = 4096;     // sequence
constexpr int DE = 1024;     // d_emb
constexpr int DH = 64;       // d_head
constexpr int QB = SS / 16;  // query blocks per batch = 256

// ---------------------------------------------------------------------------
// Kernel 0: WT[w][n][k] = W_w[k][n]   (3 x 64 x 1024 fp32, 0.75 MB total)
// ---------------------------------------------------------------------------
__global__ __launch_bounds__(256) void wt_kernel(
    const float* __restrict__ Wq, const float* __restrict__ Wk,
    const float* __restrict__ Wv, float* __restrict__ WT)
{
    const float* __restrict__ Ws[3] = {Wq, Wk, Wv};
    int g = blockIdx.x * 256 + threadIdx.x;          // 0 .. 3*DH*DE-1
    int w = g / (DH * DE);
    int r = g % (DH * DE);
    int n = r / DE;
    int k = r % DE;
    WT[g] = Ws[w][k * DH + n];
}

// ---------------------------------------------------------------------------
// Kernel 1: one wave per (64-row tile, matrix w).  Per K-chunk of 4:
//   4x b64 A loads (x rows), 4x b64 B loads (W^T), 16 WMMAs.
// w==2 (V) is stored transposed per batch: Vt[b][d][s].
// ---------------------------------------------------------------------------
__global__ __launch_bounds__(32) void qkv_kernel(
    const float* __restrict__ x, const float* __restrict__ WT,
    float* __restrict__ Q, float* __restrict__ Kq, float* __restrict__ Vt)
{
    const int  w    = blockIdx.y;                    // 0=Q, 1=K, 2=V
    const int  lane = threadIdx.x & 31;
    const int  l16  = lane & 15;
    const int  hi   = lane >> 4;
    const long rowbase = (long)blockIdx.x * 64;

    const float* Wt = WT + (long)w * DH * DE;        // [n][k]

    v8f vzero = {};
    v8f acc[4][4];                                   // [mtile][ntile]
#pragma unroll
    for (int mt = 0; mt < 4; ++mt)
#pragma unroll
        for (int t = 0; t < 4; ++t) acc[mt][t] = vzero;

    const float* xb = x + (rowbase + l16) * DE + 2 * hi;
    const float* wb = Wt + l16 * DE + 2 * hi;

    for (int kc = 0; kc < DE; kc += 4) {
        v2f a[4], bb[4];
#pragma unroll
        for (int mt = 0; mt < 4; ++mt)
            a[mt] = *(const v2f*)(xb + (long)mt * 16 * DE + kc);
#pragma unroll
        for (int t = 0; t < 4; ++t)
            bb[t] = *(const v2f*)(wb + (long)t * 16 * DE + kc);
#pragma unroll
        for (int mt = 0; mt < 4; ++mt)
#pragma unroll
            for (int t = 0; t < 4; ++t)
                acc[mt][t] = WMMA_F32(a[mt], bb[t], acc[mt][t]);
    }

    if (w < 2) {
        float* __restrict__ O = (w == 0) ? Q : Kq;
#pragma unroll
        for (int mt = 0; mt < 4; ++mt)
#pragma unroll
            for (int t = 0; t < 4; ++t)
#pragma unroll
                for (int r = 0; r < 8; ++r)
                    O[(rowbase + mt * 16 + r + 8 * hi) * DH + t * 16 + l16] =
                        acc[mt][t][r];
    } else {
#pragma unroll
        for (int mt = 0; mt < 4; ++mt)
#pragma unroll
            for (int t = 0; t < 4; ++t)
#pragma unroll
                for (int r = 0; r < 8; ++r) {
                    long row = rowbase + mt * 16 + r + 8 * hi;   // global token
                    long bI  = row >> 12;                        // / SS
                    long s   = row & (SS - 1);                   // % SS
                    Vt[bI * (long)DH * SS + (t * 16 + l16) * (long)SS + s] =
                        acc[mt][t][r];
                }
    }
}

// ---------------------------------------------------------------------------
// Kernel 2: flash attention with causal mask.  One wave per (batch, qblock).
// ---------------------------------------------------------------------------
__global__ __launch_bounds__(32) void attn_kernel(
    const float* __restrict__ Q, const float* __restrict__ Kq,
    const float* __restrict__ Vt, float* __restrict__ out)
{
    const int b    = blockIdx.x >> 8;   // / QB
    const int qb   = blockIdx.x & 255;  // % QB
    const int lane = threadIdx.x & 31;
    const int l16  = lane & 15;
    const int hi   = lane >> 4;

    const float* Qb  = Q  + (long)b * SS * DH;
    const float* Kb  = Kq + (long)b * SS * DH;
    const float* Vtb = Vt + (long)b * DH * SS;       // [d][s]

    // Preload Q^T in B-operand layout: reg j -> d = 4c + j + 2*hi, lane -> query.
    const long qrow = (long)qb * 16 + l16;
    v2f qreg[16];
#pragma unroll
    for (int c = 0; c < 16; ++c)
        qreg[c] = *(const v2f*)(Qb + qrow * DH + 4 * c + 2 * hi);

    // Per-d-row base pointers into V^T (lane -> d = t*16 + l16).
    const float* Vrow[4];
#pragma unroll
    for (int t = 0; t < 4; ++t)
        Vrow[t] = Vtb + (t * 16 + l16) * (long)SS + 2 * hi;

    v8f vzero = {};
    v8f acc[4];                                      // O^T tiles
#pragma unroll
    for (int t = 0; t < 4; ++t) acc[t] = vzero;

    float mrow = -__builtin_inff();
    float lsum = 0.0f;
    const float scale = 0.125f;                      // 1/sqrt(64)
    const int   query = qb * 16 + l16;

    for (int kb = 0; kb <= qb; ++kb) {
        // ---- S^T = K * Q^T over d = 0..63 (16 chunks of K=4) ----
        v8f st = vzero;
        const float* Kp = Kb + ((long)kb * 16 + l16) * DH + 2 * hi;
#pragma unroll
        for (int c = 0; c < 16; ++c) {
            v2f ka = *(const v2f*)(Kp + 4 * c);      // K[key][4c+klo .. +1]
            st = WMMA_F32(ka, qreg[c], st);
        }

        // ---- scale + causal mask (key = kb*16 + r + 8*hi, query on lane) ----
#pragma unroll
        for (int r = 0; r < 8; ++r) {
            int   key = kb * 16 + r + 8 * hi;
            float s   = st[r] * scale;
            st[r] = (key <= query) ? s : -__builtin_inff();
        }

        // ---- online softmax: row stats live per lane ----
        float bm = st[0];
#pragma unroll
        for (int r = 1; r < 8; ++r) bm = fmaxf(bm, st[r]);
        bm = fmaxf(bm, __shfl_xor(bm, 16, 32));      // combine key halves
        float mnew = fmaxf(mrow, bm);
        float corr = __expf(mrow - mnew);            // first block: exp(-inf)=0

        float psum = 0.0f;
#pragma unroll
        for (int r = 0; r < 8; ++r) {
            float p = __expf(st[r] - mnew);
            st[r] = p;
            psum += p;
        }
        psum += __shfl_xor(psum, 16, 32);
        lsum  = lsum * corr + psum;
        mrow  = mnew;

#pragma unroll
        for (int t = 0; t < 4; ++t)
#pragma unroll
            for (int r = 0; r < 8; ++r) acc[t][r] *= corr;

        // ---- O^T += V^T * P : P (C-layout) -> B operand via half swaps ----
#pragma unroll
        for (int c = 0; c < 4; ++c) {
            v2f bop;
#pragma unroll
            for (int j = 0; j < 2; ++j) {
                float own, oth;
                if (c < 2) {                         // n_lo in lower-half regs
                    own = st[4 * c + j];
                    oth = __shfl_xor(st[4 * c + 2 + j], 16, 32);
                    bop[j] = hi ? oth : own;
                } else {                             // n_lo lives in upper-half regs
                    int cc = c - 2;
                    oth = __shfl_xor(st[4 * cc + j], 16, 32);
                    own = st[4 * cc + 2 + j];
                    bop[j] = hi ? own : oth;
                }
            }
            // A = V^T chunk: lane -> d, reg j -> key = kb*16 + 4c + 2hi + j  (b64)
#pragma unroll
            for (int t = 0; t < 4; ++t) {
                v2f va = *(const v2f*)(Vrow[t] + (long)kb * 16 + 4 * c);
                acc[t] = WMMA_F32(va, bop, acc[t]);
            }
        }
    }

    // ---- epilogue: O = O^T^T / l ----
    float  inv = 1.0f / lsum;
    float* Ob  = out + ((long)b * SS + qb * 16 + l16) * DH;
#pragma unroll
    for (int t = 0; t < 4; ++t)
#pragma unroll
        for (int r = 0; r < 8; ++r)
            Ob[t * 16 + r + 8 * hi] = acc[t][r] * inv;
}

// ---------------------------------------------------------------------------
extern "C" void kernel_launch(void* const* d_in, const int* in_sizes, int n_in,
                              void* d_out, int out_size, void* d_ws, size_t ws_size,
                              hipStream_t stream)
{
    (void)in_sizes; (void)n_in; (void)out_size; (void)ws_size;
    const float* x  = (const float*)d_in[0];
    const float* Wq = (const float*)d_in[1];
    const float* Wk = (const float*)d_in[2];
    const float* Wv = (const float*)d_in[3];
    float* out = (float*)d_out;

    const long N = (long)BB * SS * DH;   // 1,048,576 floats (4 MB) per tensor
    float* Qw = (float*)d_ws;            // [b][s][d]
    float* Kw = Qw + N;                  // [b][s][d]
    float* Vw = Kw + N;                  // [b][d][s]  (transposed)
    float* WT = Vw + N;                  // [3][DH][DE]

    wt_kernel<<<(3 * DH * DE) / 256, 256, 0, stream>>>(Wq, Wk, Wv, WT);
    qkv_kernel<<<dim3((BB * SS) / 64, 3), 32, 0, stream>>>(x, WT, Qw, Kw, Vw);
    attn_kernel<<<BB * QB, 32, 0, stream>>>(Qw, Kw, Vw, out);
}